// GCN_5342939316732
// MI455X (gfx1250) — compile-verified
//
#include <hip/hip_runtime.h>

typedef __attribute__((ext_vector_type(2))) float v2f;
typedef __attribute__((ext_vector_type(8))) float v8f;

#define DIN 128

// ---------- degree / normalization ----------
__global__ void k_zero_u32(unsigned* __restrict__ p, int n) {
  int i = blockIdx.x * blockDim.x + threadIdx.x;
  if (i < n) p[i] = 0u;
}

__global__ void k_count_deg(const int* __restrict__ dst, unsigned* __restrict__ deg, int nE) {
  int e = blockIdx.x * blockDim.x + threadIdx.x;
  if (e < nE) atomicAdd(&deg[dst[e]], 1u);
}

__global__ void k_inv_sqrt(const unsigned* __restrict__ deg, float* __restrict__ invs, int n) {
  int i = blockIdx.x * blockDim.x + threadIdx.x;
  if (i < n) invs[i] = rsqrtf((float)deg[i] + 1.0f);   // deg + self-loop
}

// ---------- fp32 WMMA GEMM: O[M x DOUT] = A[M x 128] * W[128 x DOUT] ----------
// One wave computes a 16-row stripe across all DOUT columns.
// V_WMMA_F32_16X16X4_F32 keeps exact fp32 precision (GEMM is not the bottleneck).
template <int DOUT>
__global__ void k_gemm_wmma(const float* __restrict__ A, const float* __restrict__ W,
                            float* __restrict__ O, int M) {
  constexpr int NT = DOUT / 16;
  int wave = (int)((blockIdx.x * blockDim.x + threadIdx.x) >> 5);
  int lane = threadIdx.x & 31;
  int row0 = wave * 16;
  if (row0 >= M) return;                 // wave-uniform: EXEC stays all-ones for WMMA
  int half = lane >> 4;                  // 0: lanes 0-15, 1: lanes 16-31
  int l16  = lane & 15;

  v8f acc[NT];
  const v8f vzero = {0.f, 0.f, 0.f, 0.f, 0.f, 0.f, 0.f, 0.f};
#pragma unroll
  for (int n = 0; n < NT; n++) acc[n] = vzero;

  // A fragment: lane half selects K pair (K, K+1) vs (K+2, K+3); row = row0 + l16
  const float* arow = A + (size_t)(row0 + l16) * DIN + 2 * half;

  for (int kk = 0; kk < DIN; kk += 4) {
    v2f a;
    a.x = arow[kk];
    a.y = arow[kk + 1];
    // B fragment: row K striped across lanes; lane half selects K vs K+2
    const float* wr0 = W + (size_t)(kk + 2 * half) * DOUT + l16;
    const float* wr1 = wr0 + DOUT;
#pragma unroll
    for (int n = 0; n < NT; n++) {
      v2f b;
      b.x = wr0[n * 16];
      b.y = wr1[n * 16];
      acc[n] = __builtin_amdgcn_wmma_f32_16x16x4_f32(
          /*neg_a=*/false, a, /*neg_b=*/false, b,
          /*c_mod=*/(short)0, acc[n], /*reuse_a=*/false, /*reuse_b=*/false);
    }
  }

  // C/D layout: VGPR r holds M = r (lanes 0-15) and M = r+8 (lanes 16-31), N = l16
#pragma unroll
  for (int n = 0; n < NT; n++) {
#pragma unroll
    for (int r = 0; r < 8; r++) {
      O[(size_t)(row0 + r + 8 * half) * DOUT + n * 16 + l16] = acc[n][r];
    }
  }
}

// ---------- self-loop term + bias init: out[i,f] = h[i,f]/deg[i] + bias[f] ----------
template <int F>
__global__ void k_self_init(const float* __restrict__ h, const float* __restrict__ invs,
                            const float* __restrict__ bias, float* __restrict__ out, int nN) {
  int idx = blockIdx.x * blockDim.x + threadIdx.x;
  int i = idx / F;
  int f = idx % F;
  if (i >= nN) return;
  float is = invs[i];
  out[idx] = h[idx] * (is * is) + bias[f];
}

// ---------- edge scatter: out[dst] += h[src] * invs[src]*invs[dst] ----------
// F/4 lanes per edge, each lane owns a float4 of features -> coalesced gather,
// hardware fp32 atomics into L2 (working set is L2-resident on MI455X's 192MB L2).
template <int F>
__global__ void k_edge_scatter(const int* __restrict__ src, const int* __restrict__ dst,
                               const float* __restrict__ invs, const float* __restrict__ h,
                               float* __restrict__ out, int nE) {
  constexpr int G = F / 4;
  int idx = blockIdx.x * blockDim.x + threadIdx.x;
  int e = idx / G;
  if (e >= nE) return;
  int g = idx - e * G;
  int s = src[e];
  int d = dst[e];
  float coef = invs[s] * invs[d];
  const float4* hv4 = (const float4*)(h + (size_t)s * F);
  float4 hv = hv4[g];
  float* o = out + (size_t)d * F + (size_t)g * 4;
  unsafeAtomicAdd(o + 0, hv.x * coef);
  unsafeAtomicAdd(o + 1, hv.y * coef);
  unsafeAtomicAdd(o + 2, hv.z * coef);
  unsafeAtomicAdd(o + 3, hv.w * coef);
}

__global__ void k_relu(float* __restrict__ p, int n) {
  int i = blockIdx.x * blockDim.x + threadIdx.x;
  if (i < n) p[i] = fmaxf(p[i], 0.0f);
}

extern "C" void kernel_launch(void* const* d_in, const int* in_sizes, int n_in,
                              void* d_out, int out_size, void* d_ws, size_t ws_size,
                              hipStream_t stream) {
  const float* x  = (const float*)d_in[0];   // [N,128]
  const int*   ei = (const int*)d_in[1];     // [2,E]
  const float* W1 = (const float*)d_in[2];   // [128,128]
  const float* b1 = (const float*)d_in[3];   // [128]
  const float* W2 = (const float*)d_in[4];   // [128,64]
  const float* b2 = (const float*)d_in[5];   // [64]
  float* out = (float*)d_out;                // [N,64]

  const int nN = in_sizes[0] / 128;
  const int nE = in_sizes[1] / 2;
  const int* src = ei;        // row 0
  const int* dst = ei + nE;   // row 1

  // Workspace layout (4B units): deg[N] | invs[N] | h1[N*128] | z1[N*128] | h2[N*64]
  float* ws = (float*)d_ws;
  unsigned* deg = (unsigned*)ws;                 // N
  float* invs = ws + (size_t)nN;                 // N
  float* h1   = ws + 2 * (size_t)nN;             // N*128
  float* z1   = h1 + (size_t)nN * 128;           // N*128
  float* h2   = z1 + (size_t)nN * 128;           // N*64

  const int TB = 256;
  auto cdiv = [](long long a, int b) { return (int)((a + b - 1) / b); };

  // degrees -> invs
  k_zero_u32<<<cdiv(nN, TB), TB, 0, stream>>>(deg, nN);
  k_count_deg<<<cdiv(nE, TB), TB, 0, stream>>>(dst, deg, nE);
  k_inv_sqrt<<<cdiv(nN, TB), TB, 0, stream>>>(deg, invs, nN);

  const int nWaves = (nN + 15) / 16;           // one wave per 16-row tile
  const int gemmBlocks = (nWaves + 3) / 4;     // 4 waves (128 threads) per block

  // ---- layer 1 ----
  k_gemm_wmma<128><<<gemmBlocks, 128, 0, stream>>>(x, W1, h1, nN);
  k_self_init<128><<<cdiv((long long)nN * 128, TB), TB, 0, stream>>>(h1, invs, b1, z1, nN);
  k_edge_scatter<128><<<cdiv((long long)nE * 32, TB), TB, 0, stream>>>(src, dst, invs, h1, z1, nE);
  k_relu<<<cdiv((long long)nN * 128, TB), TB, 0, stream>>>(z1, nN * 128);

  // ---- layer 2 ----
  k_gemm_wmma<64><<<gemmBlocks, 128, 0, stream>>>(z1, W2, h2, nN);
  k_self_init<64><<<cdiv((long long)nN * 64, TB), TB, 0, stream>>>(h2, invs, b2, out, nN);
  k_edge_scatter<64><<<cdiv((long long)nE * 16, TB), TB, 0, stream>>>(src, dst, invs, h2, out, nE);
}